// DiffPathRenderer_62818191671426
// MI455X (gfx1250) — compile-verified
//
#include <hip/hip_runtime.h>

typedef __attribute__((ext_vector_type(2))) float v2f;
typedef __attribute__((ext_vector_type(8))) float v8f;

#define CANVAS 512
#define NSEG   63   // 64 points -> 63 segments
#define NSEGP  64   // padded to 4 tiles of 16

// One wave renders 16 consecutive pixels of one row against all 64 (padded)
// segments using v_wmma_f32_16x16x4_f32:
//   A (16x4): row m = [px, py, |p|^2, 1]
//   Bq(4x16): col n = [sx, sy, 0, -v.seg]      -> D = (p-v).seg  (= q)
//   Be(4x16): col n = [-2vx, -2vy, 1, |v|^2]   -> D = |p-v|^2    (= e)
// then dist^2 = e - tt*(2q - tt*d2), tt = clamp(q/(d2+1e-5), 0, 1).
__global__ __launch_bounds__(256)
void DiffPathRenderer_wmma_kernel(const float* __restrict__ traj,
                                  const float* __restrict__ thick,
                                  float* __restrict__ out) {
    __shared__ float sSx[NSEGP], sSy[NSEGP];       // segment vector
    __shared__ float sM2vx[NSEGP], sM2vy[NSEGP];   // -2*v
    __shared__ float sNegVd[NSEGP];                // -(v . seg)
    __shared__ float sV2[NSEGP];                   // |v|^2
    __shared__ float sInv[NSEGP];                  // 1/(d2 + 1e-5)
    __shared__ float sD2[NSEGP];                   // d2
    __shared__ float sOut[8 * 16];                 // per-wave staging

    const int tid = threadIdx.x;

    // ---- per-block segment precompute (threads 0..63) ----
    if (tid < NSEGP) {
        const int s = tid;
        float vx, vy, ex, ey;
        if (s < NSEG) {
            vx = traj[2 * s + 0] * (float)CANVAS;
            vy = traj[2 * s + 1] * (float)CANVAS;
            ex = traj[2 * s + 2] * (float)CANVAS;
            ey = traj[2 * s + 3] * (float)CANVAS;
        } else {
            // padding segment, far away: dist^2 ~ 2e16 never wins the min
            vx = 1.0e8f; vy = 1.0e8f; ex = vx + 1.0f; ey = vy;
        }
        const float sx = ex - vx, sy = ey - vy;
        const float d2 = sx * sx + sy * sy;
        sSx[s]    = sx;
        sSy[s]    = sy;
        sM2vx[s]  = -2.0f * vx;
        sM2vy[s]  = -2.0f * vy;
        sNegVd[s] = -(vx * sx + vy * sy);
        sV2[s]    = vx * vx + vy * vy;
        sInv[s]   = 1.0f / (d2 + 1e-5f);
        sD2[s]    = d2;
    }
    __syncthreads();

    // ---- wave / pixel mapping ----
    const int lane    = tid & 31;
    const int waveIdx = tid >> 5;
    const int w       = blockIdx.x * 8 + waveIdx;  // global wave id
    const int y       = w >> 5;                    // pixel row
    const int x0      = (w & 31) << 4;             // first pixel column
    const int h       = lane >> 4;                 // half-wave select
    const int m       = lane & 15;                 // row-in-tile / seg-in-tile

    // A-matrix per ISA 32-bit 16x4 layout:
    //  lanes 0-15 : VGPR0=K0(px), VGPR1=K1(py)   for row M=lane
    //  lanes 16-31: VGPR0=K2(p2), VGPR1=K3(1)    for row M=lane-16
    const float px = (float)(x0 + m);
    const float py = (float)y;
    v2f A;
    A.x = h ? fmaf(px, px, py * py) : px;
    A.y = h ? 1.0f : py;

    float mind2[8];
#pragma unroll
    for (int r = 0; r < 8; ++r) mind2[r] = 3.0e38f;

    const v8f zero = {};

#pragma unroll
    for (int t = 0; t < 4; ++t) {               // 4 segment tiles of 16
        const int idx = t * 16 + m;
        const float sx  = sSx[idx],   sy  = sSy[idx];
        const float nvd = sNegVd[idx], vv2 = sV2[idx];
        const float m2x = sM2vx[idx], m2y = sM2vy[idx];
        const float inv = sInv[idx],  dd2 = sD2[idx];

        // B-matrix 4x16: lanes 0-15 hold K0/K1, lanes 16-31 hold K2/K3
        v2f Bq, Be;
        Bq.x = h ? 0.0f : sx;   Bq.y = h ? nvd : sy;
        Be.x = h ? 1.0f : m2x;  Be.y = h ? vv2 : m2y;

        v8f Dq = __builtin_amdgcn_wmma_f32_16x16x4_f32(
            false, A, false, Bq, (short)0, zero, false, false);
        v8f De = __builtin_amdgcn_wmma_f32_16x16x4_f32(
            false, A, false, Be, (short)0, zero, false, false);

#pragma unroll
        for (int r = 0; r < 8; ++r) {
            const float q  = Dq[r];
            const float e  = De[r];
            const float tt = fminf(fmaxf(q * inv, 0.0f), 1.0f);
            const float u  = fmaf(-tt, dd2, 2.0f * q);  // 2q - tt*d2
            const float dd = fmaf(-tt, u, e);           // dist^2
            mind2[r] = fminf(mind2[r], dd);
        }
    }

    // ---- min across the 16-lane segment groups ----
#pragma unroll
    for (int r = 0; r < 8; ++r) {
        float v = mind2[r];
        v = fminf(v, __shfl_xor(v, 1, 32));
        v = fminf(v, __shfl_xor(v, 2, 32));
        v = fminf(v, __shfl_xor(v, 4, 32));
        v = fminf(v, __shfl_xor(v, 8, 32));
        mind2[r] = v;
    }

    const float radius = 0.5f * thick[0];
    const float invR   = 1.0f / radius;

    // D layout: VGPR r <-> pixel m = r (lanes 0-15) or 8+r (lanes 16-31)
#pragma unroll
    for (int r = 0; r < 8; ++r) {
        if (m == 0) {
            const float dist = sqrtf(mind2[r]);
            const float dark = fminf(fmaxf((radius - dist) * invR, 0.0f), 1.0f);
            sOut[waveIdx * 16 + h * 8 + r] = dark;
        }
    }
    __syncthreads();

    if (lane < 16) {
        out[y * CANVAS + x0 + lane] = sOut[waveIdx * 16 + lane];
    }
}

extern "C" void kernel_launch(void* const* d_in, const int* in_sizes, int n_in,
                              void* d_out, int out_size, void* d_ws, size_t ws_size,
                              hipStream_t stream) {
    (void)in_sizes; (void)n_in; (void)out_size; (void)d_ws; (void)ws_size;
    const float* traj  = (const float*)d_in[0];  // 64 x 2 f32, normalized
    const float* thick = (const float*)d_in[1];  // scalar f32
    float* out = (float*)d_out;                  // 512 x 512 f32

    // 16384 waves total: 512 rows x 32 sixteen-pixel chunks; 8 waves/block
    dim3 grid(2048), block(256);
    DiffPathRenderer_wmma_kernel<<<grid, block, 0, stream>>>(traj, thick, out);
}